// GOAD_35725537968402
// MI455X (gfx1250) — compile-verified
//
#include <hip/hip_runtime.h>

// MI455X / gfx1250, wave32. Fused GOAD forward:
//   pass1: X@AW via V_WMMA_F32_16X16X4_F32 with double-buffered async
//          (ASYNCcnt) LDS staging of the AW tiles -> conv chain -> rep/CE/means
//   pass2: triplet-center hinge vs batch means -> scalar loss_tc
//   pass3: loss[b] = mean_t CE + ALPHA*loss_tc

typedef __attribute__((ext_vector_type(2))) float v2f;
typedef __attribute__((ext_vector_type(8))) float v8f;
typedef int v4i_vs __attribute__((vector_size(16)));   // builtin's expected elem type

#define AS1 __attribute__((address_space(1)))
#define AS3 __attribute__((address_space(3)))

#define NB 32768
#define NF 128
#define NT 32
#define ND 16
#define NH 10

__device__ __forceinline__ float leaky(float x) { return x >= 0.0f ? x : 0.2f * x; }

// Async-copy 16B (global -> LDS), tracked by ASYNCcnt.
__device__ __forceinline__ void async_cp16(const float* g, float* l) {
    __builtin_amdgcn_global_load_async_to_lds_b128(
        (AS1 v4i_vs*)(size_t)g, (AS3 v4i_vs*)l, 0, 0);
}

// ---------------------------------------------------------------------------
// Pass 1: one wave per 16 batch rows (4 waves / block).
//   A fragments (X rows) live in 64 VGPRs for the whole kernel.
//   AW column-tile pairs (16KB, contiguous) are double-buffered into LDS with
//   async loads; WMMA chains consume tile j while tile j+1 streams in.
// ---------------------------------------------------------------------------
__global__ __launch_bounds__(128, 4) void goad_pass1(
    const float* __restrict__ X,  const float* __restrict__ AW,
    const float* __restrict__ W0, const float* __restrict__ W1,
    const float* __restrict__ W2, const float* __restrict__ W3,
    const float* __restrict__ Wh, const float* __restrict__ bh,
    float* __restrict__ rep_out,  float* __restrict__ loss_out,
    float* __restrict__ mean_sums)
{
    __shared__ float sW0[NH * ND];
    __shared__ float sW1[NH * NH], sW2[NH * NH], sW3[NH * NH];
    __shared__ float sWh[NT * NH];
    __shared__ float sbh[NT];
    __shared__ __align__(16) float sB[2][2][NF * ND];   // [buf][tile][n*16+d], 32KB
    __shared__ __align__(16) float sXT[4][2][16 * 16];  // per-wave xt tiles, 8KB
    __shared__ float sM[NT * NH];                       // block mean-sums

    const int tid  = threadIdx.x;
    const int w    = tid >> 5;    // wave in block
    const int l    = tid & 31;    // lane
    const int mrow = l & 15;      // WMMA M row (batch row) / N col (=d) for B frag
    const int grp  = l >> 4;      // WMMA lane group (selects K pair)

    // Stage weights + zero mean-sum accumulator.
    for (int i = tid; i < NH * ND; i += 128) sW0[i] = W0[i];
    for (int i = tid; i < NH * NH; i += 128) { sW1[i] = W1[i]; sW2[i] = W2[i]; sW3[i] = W3[i]; }
    for (int i = tid; i < NT * NH; i += 128) { sWh[i] = Wh[i]; sM[i] = 0.0f; }
    if (tid < NT) sbh[tid] = bh[tid];

    const int rowBase = blockIdx.x * 64 + w * 16;

    // A fragments for all 32 K-chunks, ISA 16x4 f32 layout:
    //   lane: M = l%16 ; K = (l/16)*2 + {0:.x, 1:.y} within each chunk of 4.
    // One-time strided read of this wave's 16 X rows (32x global_load_b64).
    v2f afrag[32];
    {
        const float* Xrow = X + (size_t)(rowBase + mrow) * NF + grp * 2;
        #pragma unroll
        for (int k = 0; k < 32; ++k)
            afrag[k] = *(const v2f*)(Xrow + k * 4);
    }

    // Prologue: async-stage AW tiles a=0,1 into buffer 0 (16KB contiguous).
    #pragma unroll
    for (int it = 0; it < 8; ++it) {
        const int idx4 = (it * 128 + tid) * 4;            // float index, 16B units
        async_cp16(AW + idx4, &sB[0][0][0] + idx4);
    }
    asm volatile("s_wait_asynccnt 0x0" ::: "memory");
    __syncthreads();

    float ce_acc = 0.0f;

    for (int j2 = 0; j2 < 16; ++j2) {
        const int cur = j2 & 1;
        const int jA  = j2 * 2;

        // Kick off async staging of the NEXT tile pair into the other buffer.
        if (j2 < 15) {
            const float* src = AW + (size_t)(jA + 2) * (NF * ND);
            #pragma unroll
            for (int it = 0; it < 8; ++it) {
                const int idx4 = (it * 128 + tid) * 4;
                async_cp16(src + idx4, &sB[cur ^ 1][0][0] + idx4);
            }
        }

        // Two 16-wide column tiles (a = jA, jA+1), each K=128 -> 32 WMMAs.
        // B frags from LDS: lanes 0-15 read 16 consecutive dwords (d=0..15),
        // lanes 16-31 are +32 banks -> conflict-free.
        #pragma unroll
        for (int s = 0; s < 2; ++s) {
            const float* Bt = &sB[cur][s][mrow];          // + n*16
            v8f acc = {};
            #pragma unroll
            for (int k = 0; k < 32; ++k) {
                const int n0 = k * 4 + grp * 2;
                v2f bfrag;
                bfrag.x = Bt[n0 * ND];
                bfrag.y = Bt[(n0 + 1) * ND];
                acc = __builtin_amdgcn_wmma_f32_16x16x4_f32(
                    false, afrag[k], false, bfrag, (short)0, acc, false, false);
            }
            // D layout: vgpr v, lane: M = v + 8*(l/16), N(=d) = l%16.
            #pragma unroll
            for (int v = 0; v < 8; ++v)
                sXT[w][s][(v + 8 * grp) * 16 + mrow] = acc[v];
        }

        // Scalar tail: lane handles (row = l%16, a = jA + l/16).
        const int a = jA + grp;
        float xv[ND];
        #pragma unroll
        for (int d = 0; d < ND; ++d) xv[d] = sXT[w][grp][mrow * 16 + d];

        float h0[NH];
        #pragma unroll
        for (int o = 0; o < NH; ++o) {
            float acc0 = 0.0f;
            #pragma unroll
            for (int d = 0; d < ND; ++d) acc0 = __builtin_fmaf(sW0[o * ND + d], xv[d], acc0);
            h0[o] = acc0;
        }
        float t0[NH], h1[NH];
        #pragma unroll
        for (int k = 0; k < NH; ++k) t0[k] = leaky(h0[k]);
        #pragma unroll
        for (int o = 0; o < NH; ++o) {
            float acc1 = 0.0f;
            #pragma unroll
            for (int k = 0; k < NH; ++k) acc1 = __builtin_fmaf(sW1[o * NH + k], t0[k], acc1);
            h1[o] = acc1;
        }
        #pragma unroll
        for (int k = 0; k < NH; ++k) t0[k] = leaky(h1[k]);
        #pragma unroll
        for (int o = 0; o < NH; ++o) {
            float acc2 = 0.0f;
            #pragma unroll
            for (int k = 0; k < NH; ++k) acc2 = __builtin_fmaf(sW2[o * NH + k], t0[k], acc2);
            h1[o] = acc2;                       // reuse as h2
        }
        #pragma unroll
        for (int k = 0; k < NH; ++k) t0[k] = leaky(h1[k]);
        float rep[NH];
        #pragma unroll
        for (int o = 0; o < NH; ++o) {
            float acc3 = 0.0f;
            #pragma unroll
            for (int k = 0; k < NH; ++k) acc3 = __builtin_fmaf(sW3[o * NH + k], t0[k], acc3);
            rep[o] = acc3;
        }

        // batch_rep[b][a][h] as 5x float2 (+ block-level mean-sum accumulation)
        float2* rp2 = (float2*)(rep_out + (size_t)(rowBase + mrow) * (NT * NH) + a * NH);
        #pragma unroll
        for (int i2 = 0; i2 < 5; ++i2) rp2[i2] = make_float2(rep[2 * i2], rep[2 * i2 + 1]);
        #pragma unroll
        for (int hh = 0; hh < NH; ++hh) atomicAdd(&sM[a * NH + hh], rep[hh]);

        // Head + online softmax over class dim (32), CE for class==a.
        float lr[NH];
        #pragma unroll
        for (int hh = 0; hh < NH; ++hh) lr[hh] = leaky(rep[hh]);
        float m = -3.0e38f, ssum = 0.0f, pa = 0.0f;
        for (int o = 0; o < NT; ++o) {
            float p = sbh[o];
            #pragma unroll
            for (int hh = 0; hh < NH; ++hh) p = __builtin_fmaf(sWh[o * NH + hh], lr[hh], p);
            pa = (o == a) ? p : pa;
            const float nm = fmaxf(m, p);
            ssum = ssum * __expf(m - nm) + __expf(p - nm);
            m = nm;
        }
        ce_acc += (m + __logf(ssum) - pa);

        // Own async stores landed + everyone done reading sB[cur] before the
        // next iteration overwrites it.
        asm volatile("s_wait_asynccnt 0x0" ::: "memory");
        __syncthreads();
    }

    // Lanes l and l^16 share a row and cover complementary a's: combine CE.
    const int pv = __builtin_amdgcn_ds_bpermute((l ^ 16) << 2, __float_as_int(ce_acc));
    const float ce_tot = ce_acc + __int_as_float(pv);
    if (grp == 0) loss_out[rowBase + mrow] = ce_tot * (1.0f / 32.0f);

    __syncthreads();
    for (int i = tid; i < NT * NH; i += 128) atomicAdd(&mean_sums[i], sM[i]);
}

// ---------------------------------------------------------------------------
// Pass 2: triplet-center hinge per (b,t) vs batch means; reduce to scalar.
// ---------------------------------------------------------------------------
__global__ __launch_bounds__(256) void goad_pass2(
    const float* __restrict__ rep, const float* __restrict__ mean_sums,
    float* __restrict__ tc_sum)
{
    __shared__ float sMean[NT * NH];
    __shared__ float red[256];
    const int tid = threadIdx.x;
    for (int i = tid; i < NT * NH; i += 256) sMean[i] = mean_sums[i] * (1.0f / (float)NB);
    __syncthreads();

    float acc = 0.0f;
    for (int it = 0; it < 8; ++it) {
        const int task = it * (512 * 256) + blockIdx.x * 256 + tid;   // b*32 + t
        const int b = task >> 5, t = task & 31;
        const float2* xp2 = (const float2*)(rep + (size_t)b * (NT * NH) + t * NH);
        float x[NH];
        #pragma unroll
        for (int i2 = 0; i2 < 5; ++i2) {
            const float2 v = xp2[i2];
            x[2 * i2] = v.x; x[2 * i2 + 1] = v.y;
        }
        float pos = 0.0f, neg = 3.0e38f;
        for (int c = 0; c < NT; ++c) {
            float r = 0.0f;
            #pragma unroll
            for (int h = 0; h < NH; ++h) {
                const float d = x[h] - sMean[c * NH + h];
                r = __builtin_fmaf(d, d, r);
            }
            pos = (c == t) ? r : pos;
            neg = (c == t) ? neg : fminf(neg, r);
        }
        acc += fmaxf(pos + 1.0f - neg, 0.0f);
    }

    red[tid] = acc;
    __syncthreads();
    for (int s = 128; s > 0; s >>= 1) {
        if (tid < s) red[tid] += red[tid + s];
        __syncthreads();
    }
    if (tid == 0) atomicAdd(tc_sum, red[0]);
}

// ---------------------------------------------------------------------------
// Pass 3: loss[b] = ce_mean[b] + ALPHA * loss_tc
// ---------------------------------------------------------------------------
__global__ __launch_bounds__(256) void goad_pass3(
    float* __restrict__ loss, const float* __restrict__ tc_sum)
{
    const int i = blockIdx.x * 256 + threadIdx.x;
    const float tc = tc_sum[0] * (1.0f / ((float)NB * (float)NT));
    loss[i] += 0.1f * tc;
}

extern "C" void kernel_launch(void* const* d_in, const int* in_sizes, int n_in,
                              void* d_out, int out_size, void* d_ws, size_t ws_size,
                              hipStream_t stream) {
    (void)in_sizes; (void)n_in; (void)out_size; (void)ws_size;
    const float* X  = (const float*)d_in[0];
    const float* AW = (const float*)d_in[1];
    const float* W0 = (const float*)d_in[2];
    const float* W1 = (const float*)d_in[3];
    const float* W2 = (const float*)d_in[4];
    const float* W3 = (const float*)d_in[5];
    const float* Wh = (const float*)d_in[6];
    const float* bh = (const float*)d_in[7];

    float* rep_out  = (float*)d_out;                    // [B][T][H]
    float* loss_out = rep_out + (size_t)NB * NT * NH;   // [B]
    float* ws    = (float*)d_ws;
    float* tc    = ws;        // ws[0]        : hinge sum
    float* msums = ws + 1;    // ws[1..320]   : mean sums [T][H]

    (void)hipMemsetAsync(d_ws, 0, (1 + NT * NH) * sizeof(float), stream);
    goad_pass1<<<NB / 64, 128, 0, stream>>>(X, AW, W0, W1, W2, W3, Wh, bh,
                                            rep_out, loss_out, msums);
    goad_pass2<<<512, 256, 0, stream>>>(rep_out, msums, tc);
    goad_pass3<<<NB / 256, 256, 0, stream>>>(loss_out, tc);
}